// EquivariantNodeConv_69715909149146
// MI455X (gfx1250) — compile-verified
//
#include <hip/hip_runtime.h>
#include <cmath>
#include <complex>
#include <vector>
#include <algorithm>

// ---------------------------------------------------------------------------
// Device types
// ---------------------------------------------------------------------------
typedef float v8f __attribute__((ext_vector_type(8)));
typedef float v2f __attribute__((ext_vector_type(2)));

struct TPConsts {
  float C110[9];   // [i*3+j] = w3j(1,1,0)[i,j,0]
  float C011[9];   // [j*3+k] = w3j(0,1,1)[0,j,k]
  float C101[9];   // [i*3+k] = w3j(1,0,1)[i,0,k]
  float C121[45];  // [(i*5+j)*3+k] = w3j(1,2,1)[i,j,k]
  float scale0;    // a0 / sqrt(num_neighbors)
  float scale1;    // a1 / sqrt(num_neighbors)
};

#define EPB 256  // edges per block (8 waves x 32)

__global__ void zero_kernel(float* __restrict__ out, int n) {
  int i = blockIdx.x * blockDim.x + threadIdx.x;
  if (i < n) out[i] = 0.0f;
}

#if __has_builtin(__builtin_amdgcn_wmma_f32_16x16x4_f32)
#define HAVE_WMMA_F32X4 1
#else
#define HAVE_WMMA_F32X4 0
#endif

__launch_bounds__(EPB, 1)
__global__ void equi_conv_kernel(const float* __restrict__ f_in,
                                 const int* __restrict__ edge_index,
                                 const float* __restrict__ pos,
                                 const float* __restrict__ W1,
                                 const float* __restrict__ W2,
                                 float* __restrict__ f_out,
                                 int E, TPConsts K)
{
  __shared__ __align__(16) float W1s[10 * 16];     // row-major [basis][t]
  __shared__ __align__(16) float W2T[320 * 16];    // transposed [col][t]
  __shared__ __align__(16) float hbuf[EPB * 16];   // [local edge][t] (pre-scaled h)
#if HAVE_WMMA_F32X4
  __shared__ __align__(16) float wbuf[8][32 * 16]; // per-wave [local edge][col-in-tile]
#endif

  const int tid  = threadIdx.x;
  const int wave = tid >> 5;
  const int lane = tid & 31;
  const int hi   = lane >> 4;
  const int lo   = lane & 15;
  (void)wave; (void)hi; (void)lo;

  // Stage shared weights
  for (int i = tid; i < 160; i += EPB) W1s[i] = W1[i];
  for (int i = tid; i < 16 * 320; i += EPB) {
    int t = i / 320;
    int c = i - t * 320;
    W2T[c * 16 + t] = W2[i];
  }
  __syncthreads();

  // ------------------------- per-edge scalar phase -------------------------
  const int  e     = blockIdx.x * EPB + tid;
  const bool valid = (e < E);
  const int  ec    = valid ? e : (E - 1);
  const int  row   = edge_index[ec];
  const int  colN  = edge_index[E + ec];

  float vx = pos[row * 3 + 0] - pos[colN * 3 + 0];
  float vy = pos[row * 3 + 1] - pos[colN * 3 + 1];
  float vz = pos[row * 3 + 2] - pos[colN * 3 + 2];
  float r  = sqrtf(vx * vx + vy * vy + vz * vz);
  float inv = 1.0f / fmaxf(r, 1e-12f);
  float ux = vx * inv, uy = vy * inv, uz = vz * inv;

  const float SQ3 = 1.7320508075688772f;
  const float S15 = 3.8729833462074170f;
  const float S5  = 2.2360679774997896f;
  float sh1[3] = { SQ3 * uy, SQ3 * uz, SQ3 * ux };
  float sh2[5] = { S15 * ux * uy, S15 * uy * uz, 0.5f * S5 * (3.0f * uz * uz - 1.0f),
                   S15 * ux * uz, 0.5f * S15 * (ux * ux - uy * uy) };

  // Radial MLP layer 1: h = sqrt(2)*relu(emb @ W1); fold 1/sqrt(HIDDEN)=0.25
  float hreg[16];
  #pragma unroll
  for (int t = 0; t < 16; ++t) hreg[t] = 0.0f;
  const float step = 3.5f / 11.0f;
  #pragma unroll
  for (int i = 0; i < 10; ++i) {
    float v   = (3.5f * (float)(i + 1)) / 11.0f;
    float d   = (r - v) / step;
    float den = 1.0f - d * d;
    float u   = (den > 1e-6f) ? expf(1.0f - 1.0f / den) : 0.0f;
    float ei  = 1.14136f * u;   // sqrt(NUM_BASIS) cancels with /sqrt(NUM_BASIS)
    #pragma unroll
    for (int t = 0; t < 16; ++t) hreg[t] += ei * W1s[i * 16 + t];
  }
  #pragma unroll
  for (int t = 0; t < 16; ++t)
    hreg[t] = 1.41421356237f * fmaxf(hreg[t], 0.0f) * 0.25f;

  #pragma unroll
  for (int t = 0; t < 16; ++t) hbuf[tid * 16 + t] = hreg[t];
  __syncthreads();

  // Gather node feature x = f_in[row] (32 floats)
  float xv[32];
  {
    const float4* xp = (const float4*)(f_in + (size_t)row * 32);
    #pragma unroll
    for (int q = 0; q < 8; ++q) {
      float4 t4 = xp[q];
      xv[q * 4 + 0] = t4.x; xv[q * 4 + 1] = t4.y;
      xv[q * 4 + 2] = t4.z; xv[q * 4 + 3] = t4.w;
    }
  }
  float x0[8];
  #pragma unroll
  for (int u = 0; u < 8; ++u) x0[u] = xv[u];

  // vB[u] = sum_{i,j} x1[u,i]*sh1[j]*C110[i,j,0]
  float b1[3];
  #pragma unroll
  for (int i = 0; i < 3; ++i)
    b1[i] = sh1[0] * K.C110[i * 3 + 0] + sh1[1] * K.C110[i * 3 + 1] + sh1[2] * K.C110[i * 3 + 2];
  float vB[8];
  #pragma unroll
  for (int u = 0; u < 8; ++u)
    vB[u] = xv[8 + u * 3 + 0] * b1[0] + xv[8 + u * 3 + 1] * b1[1] + xv[8 + u * 3 + 2] * b1[2];

  // s1c[k] = sum_j sh1[j]*C011[0,j,k]
  float s1c[3];
  #pragma unroll
  for (int k = 0; k < 3; ++k)
    s1c[k] = sh1[0] * K.C011[0 * 3 + k] + sh1[1] * K.C011[1 * 3 + k] + sh1[2] * K.C011[2 * 3 + k];

  // d1v[u][k] = sum_i x1[u,i]*C101[i,0,k]   (sh0 == 1)
  float d1v[8][3];
  #pragma unroll
  for (int u = 0; u < 8; ++u)
    #pragma unroll
    for (int k = 0; k < 3; ++k)
      d1v[u][k] = xv[8 + u * 3 + 0] * K.C101[0 * 3 + k] +
                  xv[8 + u * 3 + 1] * K.C101[1 * 3 + k] +
                  xv[8 + u * 3 + 2] * K.C101[2 * 3 + k];

  // tEv[u][k] = sum_{i,j} x1[u,i]*sh2[j]*C121[i,j,k]
  float gE[3][3];
  #pragma unroll
  for (int i = 0; i < 3; ++i)
    #pragma unroll
    for (int k = 0; k < 3; ++k) {
      float a = 0.0f;
      #pragma unroll
      for (int j = 0; j < 5; ++j) a += sh2[j] * K.C121[(i * 5 + j) * 3 + k];
      gE[i][k] = a;
    }
  float tEv[8][3];
  #pragma unroll
  for (int u = 0; u < 8; ++u)
    #pragma unroll
    for (int k = 0; k < 3; ++k)
      tEv[u][k] = xv[8 + u * 3 + 0] * gE[0][k] +
                  xv[8 + u * 3 + 1] * gE[1][k] +
                  xv[8 + u * 3 + 2] * gE[2][k];

  float out0[8];
  float out1[8][3];
  #pragma unroll
  for (int w = 0; w < 8; ++w) {
    out0[w] = 0.0f;
    out1[w][0] = 0.0f; out1[w][1] = 0.0f; out1[w][2] = 0.0f;
  }

  // --------------------- WMMA GEMM:  w = h @ W2  (E x 320) -----------------
#if HAVE_WMMA_F32X4
  // A-tiles (16 edges x K=16), K split into 4 chunks of 4 per the 16x16x4 layout:
  // lane 0-15 hold K = k0,k0+1 ; lane 16-31 hold K = k0+2,k0+3
  v2f Amat[2][4];
  #pragma unroll
  for (int m = 0; m < 2; ++m)
    #pragma unroll
    for (int c = 0; c < 4; ++c)
      Amat[m][c] = *(const v2f*)&hbuf[(wave * 32 + m * 16 + lo) * 16 + c * 4 + 2 * hi];
#endif

  // Outer loop over the five weight blocks (wA,wB,wC,wD,wE) is a REAL loop
  // (uniform scalar trip) to keep the live WMMA pipeline and VGPR count small;
  // the 4 sub-tiles inside are unrolled so all per-edge arrays stay statically
  // indexed (no scratch spills).
  #pragma unroll 1
  for (int blk = 0; blk < 5; ++blk) {
    #pragma unroll
    for (int sub = 0; sub < 4; ++sub) {
      const int nb = blk * 4 + sub;   // 16-column tile index (0..19)
      float wt[16];
#if HAVE_WMMA_F32X4
      v2f Bmat[4];
      #pragma unroll
      for (int c = 0; c < 4; ++c)
        Bmat[c] = *(const v2f*)&W2T[(nb * 16 + lo) * 16 + c * 4 + 2 * hi];
      #pragma unroll
      for (int m = 0; m < 2; ++m) {
        v8f acc = {};
        #pragma unroll
        for (int c = 0; c < 4; ++c)
          acc = __builtin_amdgcn_wmma_f32_16x16x4_f32(false, Amat[m][c], false, Bmat[c],
                                                      (short)0, acc, false, false);
        // D layout: VGPR r, lanes 0-15 -> M=r, lanes 16-31 -> M=r+8 ; N = lo
        #pragma unroll
        for (int rr = 0; rr < 8; ++rr)
          wbuf[wave][(m * 16 + rr + 8 * hi) * 16 + lo] = acc[rr];
      }
      {
        const float4* wp = (const float4*)&wbuf[wave][lane * 16];
        #pragma unroll
        for (int q = 0; q < 4; ++q) {
          float4 t4 = wp[q];
          wt[q * 4 + 0] = t4.x; wt[q * 4 + 1] = t4.y;
          wt[q * 4 + 2] = t4.z; wt[q * 4 + 3] = t4.w;
        }
      }
#else
      // VALU fallback (builtin unavailable): per-lane dot products
      #pragma unroll
      for (int c = 0; c < 16; ++c) {
        float a = 0.0f;
        #pragma unroll
        for (int t = 0; t < 16; ++t) a += hreg[t] * W2T[(nb * 16 + c) * 16 + t];
        wt[c] = a;
      }
#endif
      // ------------- tensor-product contraction for these 16 columns -------
      // column = 64*blk + 8*u + w'   with u = sub*2 + hh (compile-time)
      #pragma unroll
      for (int hh = 0; hh < 2; ++hh) {
        const int u = sub * 2 + hh;
        const float* wv = &wt[hh * 8];
        if (blk == 0) {            // wA : out0 += wA[u,w]*x0[u]   (sh0 = 1)
          #pragma unroll
          for (int w = 0; w < 8; ++w) out0[w] += wv[w] * x0[u];
        } else if (blk == 1) {     // wB : out0 += wB[u,w]*vB[u]
          #pragma unroll
          for (int w = 0; w < 8; ++w) out0[w] += wv[w] * vB[u];
        } else if (blk == 2) {     // wC : out1[w,k] += wC[u,w]*x0[u]*s1c[k]
          #pragma unroll
          for (int w = 0; w < 8; ++w) {
            float f = wv[w] * x0[u];
            out1[w][0] += f * s1c[0];
            out1[w][1] += f * s1c[1];
            out1[w][2] += f * s1c[2];
          }
        } else if (blk == 3) {     // wD : out1[w,k] += wD[u,w]*d1v[u][k]
          #pragma unroll
          for (int w = 0; w < 8; ++w) {
            out1[w][0] += wv[w] * d1v[u][0];
            out1[w][1] += wv[w] * d1v[u][1];
            out1[w][2] += wv[w] * d1v[u][2];
          }
        } else {                   // wE : out1[w,k] += wE[u,w]*tEv[u][k]
          #pragma unroll
          for (int w = 0; w < 8; ++w) {
            out1[w][0] += wv[w] * tEv[u][0];
            out1[w][1] += wv[w] * tEv[u][1];
            out1[w][2] += wv[w] * tEv[u][2];
          }
        }
      }
    }
  }

  // ------------------------- scatter-add to f_out --------------------------
  if (valid) {
    float* op = f_out + (size_t)colN * 32;
    #pragma unroll
    for (int w = 0; w < 8; ++w)
      __hip_atomic_fetch_add(op + w, out0[w] * K.scale0,
                             __ATOMIC_RELAXED, __HIP_MEMORY_SCOPE_AGENT);
    #pragma unroll
    for (int w = 0; w < 8; ++w)
      #pragma unroll
      for (int k = 0; k < 3; ++k)
        __hip_atomic_fetch_add(op + 8 + w * 3 + k, out1[w][k] * K.scale1,
                               __ATOMIC_RELAXED, __HIP_MEMORY_SCOPE_AGENT);
  }
}

// ---------------------------------------------------------------------------
// Host: exact Wigner 3j / change-of-basis constants (mirrors reference math)
// ---------------------------------------------------------------------------
namespace w3jhost {
typedef std::complex<double> cd;

static double fct(int n) { double r = 1.0; for (int i = 2; i <= n; ++i) r *= (double)i; return r; }

static void su2_cg(int j1, int j2, int j3, std::vector<double>& C) {
  int d2 = 2 * j2 + 1, d3 = 2 * j3 + 1;
  C.assign((2 * j1 + 1) * d2 * d3, 0.0);
  for (int m1 = -j1; m1 <= j1; ++m1)
    for (int m2 = -j2; m2 <= j2; ++m2) {
      int m3 = m1 + m2;
      if (m3 < -j3 || m3 > j3) continue;
      double pref = std::sqrt((2.0 * j3 + 1.0) * fct(j1 + j2 - j3) * fct(j1 - j2 + j3) *
                              fct(-j1 + j2 + j3) / fct(j1 + j2 + j3 + 1));
      pref *= std::sqrt(fct(j3 + m3) * fct(j3 - m3) * fct(j1 - m1) * fct(j1 + m1) *
                        fct(j2 - m2) * fct(j2 + m2));
      double s = 0.0;
      int vlo = std::max(0, std::max(j2 - j3 - m1, j1 - j3 + m2));
      int vhi = std::min(j1 + j2 - j3, std::min(j1 - m1, j2 + m2));
      for (int v = vlo; v <= vhi; ++v)
        s += ((v & 1) ? -1.0 : 1.0) /
             (fct(v) * fct(j1 + j2 - j3 - v) * fct(j1 - m1 - v) * fct(j2 + m2 - v) *
              fct(j3 - j2 + m1 + v) * fct(j3 - j1 - m2 + v));
      C[((j1 + m1) * d2 + (j2 + m2)) * d3 + (j3 + m3)] = pref * s;
    }
}

static void qmat(int l, std::vector<cd>& q) {
  int d = 2 * l + 1;
  q.assign(d * d, cd(0, 0));
  double is2 = 1.0 / std::sqrt(2.0);
  for (int m = -l; m < 0; ++m) {
    q[(l + m) * d + (l - m)] = cd(is2, 0);
    q[(l + m) * d + (l + m)] = cd(0, -is2);
  }
  q[l * d + l] = cd(1, 0);
  for (int m = 1; m <= l; ++m) {
    double sg = (m & 1) ? -1.0 : 1.0;
    q[(l + m) * d + (l + m)] = cd(sg * is2, 0);
    q[(l + m) * d + (l - m)] = cd(0, sg * is2);
  }
  cd ph(1, 0), mi(0, -1);
  for (int i = 0; i < l; ++i) ph *= mi;
  for (size_t i = 0; i < q.size(); ++i) q[i] *= ph;
}

static void w3j(int l1, int l2, int l3, float* out) {
  int d1 = 2 * l1 + 1, d2 = 2 * l2 + 1, d3 = 2 * l3 + 1;
  std::vector<double> C;
  su2_cg(l1, l2, l3, C);
  std::vector<cd> q1, q2, q3;
  qmat(l1, q1); qmat(l2, q2); qmat(l3, q3);
  std::vector<cd> A(d1 * d2 * d3, cd(0, 0));
  for (int j = 0; j < d1; ++j)
    for (int l = 0; l < d2; ++l)
      for (int n = 0; n < d3; ++n) {
        cd acc(0, 0);
        for (int i = 0; i < d1; ++i)
          for (int k = 0; k < d2; ++k)
            for (int m = 0; m < d3; ++m)
              acc += q1[i * d1 + j] * q2[k * d2 + l] * std::conj(q3[m * d3 + n]) *
                     C[(i * d2 + k) * d3 + m];
        A[(j * d2 + l) * d3 + n] = acc;
      }
  double nr = 0, ni = 0;
  for (size_t i = 0; i < A.size(); ++i) {
    nr += A[i].real() * A[i].real();
    ni += A[i].imag() * A[i].imag();
  }
  bool useRe = std::sqrt(nr) >= std::sqrt(ni);
  double nrm = std::sqrt(useRe ? nr : ni);
  for (size_t i = 0; i < A.size(); ++i)
    out[i] = (float)((useRe ? A[i].real() : A[i].imag()) / nrm);
}
}  // namespace w3jhost

// ---------------------------------------------------------------------------
// Entry point
// ---------------------------------------------------------------------------
extern "C" void kernel_launch(void* const* d_in, const int* in_sizes, int n_in,
                              void* d_out, int out_size, void* d_ws, size_t ws_size,
                              hipStream_t stream) {
  const float* f_in       = (const float*)d_in[0];
  const int*   edge_index = (const int*)d_in[1];
  const float* pos        = (const float*)d_in[2];
  const float* W1         = (const float*)d_in[3];
  const float* W2         = (const float*)d_in[4];
  float*       out        = (float*)d_out;

  const int E = in_sizes[1] / 2;        // edge_index is (2, E)
  const int N = in_sizes[0] / 32;       // f_in is (N, 4*MUL)

  TPConsts K;
  w3jhost::w3j(1, 1, 0, K.C110);        // (3,3,1) -> [i*3+j]
  w3jhost::w3j(0, 1, 1, K.C011);        // (1,3,3) -> [j*3+k]
  w3jhost::w3j(1, 0, 1, K.C101);        // (3,1,3) -> [i*3+k]
  w3jhost::w3j(1, 2, 1, K.C121);        // (3,5,3) -> [(i*5+j)*3+k]

  const double numNeigh = (double)E / (double)N;
  const double invN = 1.0 / std::sqrt(numNeigh);
  K.scale0 = (float)(std::sqrt(1.0 / 16.0) * invN);  // a0 = sqrt(1/(2*MUL))
  K.scale1 = (float)(std::sqrt(1.0 / 8.0) * invN);   // a1 = sqrt(3/(3*MUL))

  zero_kernel<<<(out_size + 255) / 256, 256, 0, stream>>>(out, out_size);

  const int blocks = (E + EPB - 1) / EPB;
  equi_conv_kernel<<<blocks, EPB, 0, stream>>>(f_in, edge_index, pos, W1, W2, out, E, K);
}